// MegaMerge_77283641524594
// MI455X (gfx1250) — compile-verified
//
#include <hip/hip_runtime.h>
#include <cstdint>
#include <cstddef>

// MegaMerge: out [4D, T] from h [T, D], c2q [D, T], q2c [D, T]:
//   out[0D+d][t] = h[t][d]
//   out[1D+d][t] = c2q[d][t]
//   out[2D+d][t] = h[t][d]*c2q[d][t]
//   out[3D+d][t] = h[t][d]*q2c[d][t]
//
// Pure bandwidth problem (~115 MB -> ~5 us @ 23.3 TB/s). The h transpose is
// done by the CDNA5 async global->LDS DMA (per-lane LDS scatter == free
// transpose, ASYNCcnt-tracked). Everything else is 128-bit, nontemporal,
// fully coalesced along t (wave32 x b128 = 512 B per instruction).

#define T_CTX  500
#define D_FEAT 8192
#define TD     32    // d-tile
#define TT     128   // t-tile (32 lanes x float4)
#define LDSS   132   // LDS row stride in words: 528 B, 16B-aligned rows

typedef float v4f __attribute__((ext_vector_type(4)));

__global__ __launch_bounds__(256) void MegaMerge_77283641524594_kernel(
    const float* __restrict__ h,     // [T, D]
    const float* __restrict__ c2q,   // [D, T]
    const float* __restrict__ q2c,   // [D, T]
    float* __restrict__ out)         // [4D, T]
{
    __shared__ float tile[TD * LDSS];   // h-tile TRANSPOSED: [d_local][t_local]

    const int lane = threadIdx.x & 31;  // wave32 lane
    const int wave = threadIdx.x >> 5;  // 0..7
    const int d0   = blockIdx.x * TD;   // always in range (D % 32 == 0)
    const int t0   = blockIdx.y * TT;

    // 32-bit LDS byte address for the DMA destination (flat LDS aperture:
    // addr[31:0] == LDS offset).
    const unsigned lds_base = (unsigned)(uintptr_t)(void*)tile;

    // ---- Phase 1: async-DMA h tile into LDS, transposed on the fly -------
    // Per instruction: lane varies d -> 128 B contiguous global read; LDS
    // side scatters to [d][t]. 16 rows per wave covers t_local 0..127.
    const float* gbase = h + (size_t)(d0 + lane);
#pragma unroll
    for (int k = 0; k < 16; ++k) {
        const int r = wave * 16 + k;               // t_local 0..127
        int tg = t0 + r;
        if (tg >= T_CTX) tg = T_CTX - 1;           // clamp; duplicate is harmless
        const float* gp = gbase + (size_t)tg * D_FEAT;
        const unsigned lds_off = lds_base + 4u * (unsigned)(lane * LDSS + r);
        asm volatile("global_load_async_to_lds_b32 %0, %1, off"
                     :
                     : "v"(lds_off), "v"(gp)
                     : "memory");
    }
    // Wait for this wave's DMA (ASYNCcnt), then publish LDS to the block.
    asm volatile("s_wait_asynccnt 0" ::: "memory");
    __syncthreads();

    // ---- Phase 2: 128-bit streamed compute + stores -----------------------
    // lane covers t = t0 + 4*lane .. +3; T % 4 == 0 so validity is per-lane
    // all-or-nothing, and every row pitch (2000 B) keeps float4 alignment.
    const int tg = t0 + lane * 4;
    if (tg < T_CTX) {
        const size_t blk = (size_t)D_FEAT * T_CTX;
#pragma unroll
        for (int k = 0; k < 4; ++k) {
            const int dl = wave + 8 * k;           // wave-uniform d_local
            const int dg = d0 + dl;
            const v4f hv = *(const v4f*)&tile[dl * LDSS + lane * 4]; // ds_load_b128
            const size_t row = (size_t)dg * T_CTX + (size_t)tg;
            const v4f cv = __builtin_nontemporal_load((const v4f*)(c2q + row));
            const v4f qv = __builtin_nontemporal_load((const v4f*)(q2c + row));
            float* o = out + row;
            __builtin_nontemporal_store(hv,      (v4f*)(o));           // H.T
            __builtin_nontemporal_store(cv,      (v4f*)(o + blk));     // c2q
            __builtin_nontemporal_store(hv * cv, (v4f*)(o + 2 * blk)); // (H*C).T
            __builtin_nontemporal_store(hv * qv, (v4f*)(o + 3 * blk)); // (H*Q).T
        }
    }
}

extern "C" void kernel_launch(void* const* d_in, const int* in_sizes, int n_in,
                              void* d_out, int out_size, void* d_ws, size_t ws_size,
                              hipStream_t stream) {
    const float* h   = (const float*)d_in[0];   // [1, T, D] flat
    const float* c2q = (const float*)d_in[1];   // [D, T]
    const float* q2c = (const float*)d_in[2];   // [D, T]
    float* out = (float*)d_out;                 // [4D, T]

    dim3 grid(D_FEAT / TD, (T_CTX + TT - 1) / TT);   // 256 x 4 blocks
    dim3 block(256);                                  // 8 wave32 waves
    hipLaunchKernelGGL(MegaMerge_77283641524594_kernel, grid, block, 0, stream,
                       h, c2q, q2c, out);
}